// RNN_54528904790514
// MI455X (gfx1250) — compile-verified
//
#include <hip/hip_runtime.h>
#include <math.h>

typedef __attribute__((ext_vector_type(2))) float v2f;
typedef __attribute__((ext_vector_type(8))) float v8f;

#define B_   256
#define S_   512
#define V_   50000
#define E_   300
#define H_   50
#define O_   4
#define HP   64      // H padded to 4 N-tiles of 16 (storage layout)
#define KT   13      // k-steps in recurrence: 13*4 = 52 >= H (rows 50..51 are zero)
#define KC   152     // K-chunk for kernel1 LDS staging (multiple of 4)

__device__ __forceinline__ float fast_tanh(float x) {
    // tanh(x) = 1 - 2/(exp(2x)+1); exact limits at +-inf, v_exp_f32 path
    float e = __expf(2.0f * x);
    return 1.0f - 2.0f / (e + 1.0f);
}

// ---------------------------------------------------------------------------
// Kernel 1: proj[v][n] = emb_table[v,:] @ W_ih^T[:,n] + (b_ih[n]+b_hh[n])
//           n in [0,64), zero for n >= 50.  V_WMMA_F32_16X16X4_F32.
// Grid: V/16 blocks, 128 threads (4 waves; wave = N-tile).
// ---------------------------------------------------------------------------
__global__ __launch_bounds__(128) void proj_kernel(
    const float* __restrict__ emb, const float* __restrict__ W_ih,
    const float* __restrict__ b_ih, const float* __restrict__ b_hh,
    float* __restrict__ proj)
{
    __shared__ float ldsB[HP][KC + 9];     // [n][k] staging, 41 KB -> ~7 blocks/WGP

    const int tid  = threadIdx.x;
    const int wave = tid >> 5;             // N-tile 0..3
    const int lane = tid & 31;
    const int half = lane >> 4;            // K-pair selector (ISA A/B layout)
    const int col  = lane & 15;
    const int nglb = wave * 16 + col;      // global h column (0..63)
    const int vbase = blockIdx.x * 16;
    const int arow  = vbase + (lane & 15); // A row: M = lane%16 (both halves)
    const float* __restrict__ aptr = emb + (size_t)arow * E_;

    const float bias = (nglb < H_) ? (b_ih[nglb] + b_hh[nglb]) : 0.0f;

    v8f c = {};
    for (int kc = 0; kc < E_; kc += KC) {
        const int klen = (E_ - kc < KC) ? (E_ - kc) : KC;   // 152 then 148
        __syncthreads();
        for (int i = tid; i < HP * klen; i += 128) {
            int nn = i / klen, k = i % klen;
            ldsB[nn][k] = (nn < H_) ? W_ih[nn * E_ + kc + k] : 0.0f;
        }
        __syncthreads();
        for (int kk = 0; kk < klen / 4; ++kk) {
            const int k0 = kk * 4 + half * 2;
            v2f a, b;
            a.x = aptr[kc + k0];
            a.y = aptr[kc + k0 + 1];
            b.x = ldsB[nglb][k0];
            b.y = ldsB[nglb][k0 + 1];
            c = __builtin_amdgcn_wmma_f32_16x16x4_f32(
                    false, a, false, b, (short)0, c, false, false);
        }
    }
#pragma unroll
    for (int r = 0; r < 8; ++r) {
        int v = vbase + r + 8 * half;            // C/D: M = r + 8*half, N = col
        proj[(size_t)v * HP + nglb] = c[r] + bias;
    }
}

// ---------------------------------------------------------------------------
// Kernel 2: 512-step tanh recurrence + FC head.
// Grid: B/16 blocks, 128 threads (4 waves = 4 N-tiles, one wave per SIMD32).
//  - W_hh^T B-fragments pinned in VGPRs (only KT=13 k-steps: K trimmed to 52)
//  - h double-buffered in LDS, one barrier/step
//  - x_t gather prefetched one step ahead (overlaps L2 latency with WMMAs)
//  - 4 independent WMMA accumulator chains (4/3/3/3) for per-SIMD ILP
// ---------------------------------------------------------------------------
__global__ __launch_bounds__(128) void rnn_kernel(
    const int*   __restrict__ tokens, const float* __restrict__ W_hh,
    const float* __restrict__ proj,
    const float* __restrict__ W_fc,  const float* __restrict__ b_fc,
    float* __restrict__ out)
{
    __shared__ float ldsW[HP][HP + 8];        // [n][k] = W_hh[n][k], zero-padded
    __shared__ float ldsH[2][16][HP + 4];     // double-buffered h tile
    __shared__ int   ldsT[16][S_];            // this block's tokens

    const int tid   = threadIdx.x;
    const int bbase = blockIdx.x * 16;

    for (int i = tid; i < HP * (HP + 8); i += 128) (&ldsW[0][0])[i] = 0.0f;
    for (int i = tid; i < 2 * 16 * (HP + 4); i += 128) (&ldsH[0][0][0])[i] = 0.0f;
    __syncthreads();
    for (int i = tid; i < H_ * H_; i += 128) {
        int n = i / H_, k = i % H_;
        ldsW[n][k] = W_hh[i];
    }
    for (int i = tid; i < 16 * S_; i += 128) {
        int m = i >> 9, s = i & (S_ - 1);
        ldsT[m][s] = tokens[(size_t)(bbase + m) * S_ + s];
    }
    __syncthreads();

    const int wave = tid >> 5;
    const int lane = tid & 31;
    const int half = lane >> 4;
    const int col  = lane & 15;
    const int n    = wave * 16 + col;   // this lane's h column
    const int m    = lane & 15;         // A row (M = lane%16)

    // Preload B operand (W_hh^T fragments) into registers: KT k-steps x v2f
    v2f bw[KT];
#pragma unroll
    for (int kk = 0; kk < KT; ++kk) {
        const int k0 = kk * 4 + half * 2;
        bw[kk].x = ldsW[n][k0];
        bw[kk].y = ldsW[n][k0 + 1];
    }

    // x_t gather for step 0
    v8f cx;
#pragma unroll
    for (int r = 0; r < 8; ++r) {
        const int tok = ldsT[r + 8 * half][0];
        cx[r] = proj[(size_t)tok * HP + n];
    }

    int cur = 0;
    for (int s = 0; s < S_; ++s) {
        // ---- prefetch next step's x_t (independent of h; hides L2 latency)
        const int sp = (s + 1) & (S_ - 1);
        v8f cn;
#pragma unroll
        for (int r = 0; r < 8; ++r) {
            const int tok = ldsT[r + 8 * half][sp];
            cn[r] = proj[(size_t)tok * HP + n];
        }

        // ---- A operand: current h from LDS (K = 0..51; cols 50/51 stay zero)
        v2f a[KT];
#pragma unroll
        for (int kk = 0; kk < KT; ++kk) {
            const int k0 = kk * 4 + half * 2;
            a[kk].x = ldsH[cur][m][k0];
            a[kk].y = ldsH[cur][m][k0 + 1];
        }

        // ---- 4 independent accumulation chains (depth 4/3/3/3)
        v8f acc0 = cx, acc1 = {}, acc2 = {}, acc3 = {};
#pragma unroll
        for (int j = 0; j < 3; ++j) {
            acc0 = __builtin_amdgcn_wmma_f32_16x16x4_f32(
                       false, a[4*j+0], false, bw[4*j+0], (short)0, acc0, false, false);
            acc1 = __builtin_amdgcn_wmma_f32_16x16x4_f32(
                       false, a[4*j+1], false, bw[4*j+1], (short)0, acc1, false, false);
            acc2 = __builtin_amdgcn_wmma_f32_16x16x4_f32(
                       false, a[4*j+2], false, bw[4*j+2], (short)0, acc2, false, false);
            acc3 = __builtin_amdgcn_wmma_f32_16x16x4_f32(
                       false, a[4*j+3], false, bw[4*j+3], (short)0, acc3, false, false);
        }
        acc0 = __builtin_amdgcn_wmma_f32_16x16x4_f32(
                   false, a[12], false, bw[12], (short)0, acc0, false, false);

        // ---- merge, activation, publish h to the other buffer
        const int nxt = cur ^ 1;
#pragma unroll
        for (int r = 0; r < 8; ++r) {
            float v = (acc0[r] + acc1[r]) + (acc2[r] + acc3[r]);
            ldsH[nxt][r + 8 * half][n] = fast_tanh(v);
        }
        cur = nxt;
        cx  = cn;
        __syncthreads();   // single barrier per step (double buffering)
    }

    // FC head: 16 rows x 4 outputs handled by threads 0..63
    if (tid < 16 * O_) {
        const int bl = tid >> 2, o = tid & 3;
        float acc = b_fc[o];
        for (int k = 0; k < H_; ++k)
            acc += ldsH[cur][bl][k] * W_fc[o * H_ + k];
        out[(bbase + bl) * O_ + o] = acc;
    }
}

// ---------------------------------------------------------------------------
extern "C" void kernel_launch(void* const* d_in, const int* in_sizes, int n_in,
                              void* d_out, int out_size, void* d_ws, size_t ws_size,
                              hipStream_t stream) {
    const int*   tokens = (const int*)  d_in[0];
    const float* emb    = (const float*)d_in[1];
    const float* W_ih   = (const float*)d_in[2];
    const float* W_hh   = (const float*)d_in[3];
    const float* b_ih   = (const float*)d_in[4];
    const float* b_hh   = (const float*)d_in[5];
    const float* W_fc   = (const float*)d_in[6];
    const float* b_fc   = (const float*)d_in[7];
    float* out  = (float*)d_out;
    float* proj = (float*)d_ws;                 // V_*HP*4 = 12.8 MB scratch

    proj_kernel<<<V_ / 16, 128, 0, stream>>>(emb, W_ih, b_ih, b_hh, proj);
    rnn_kernel<<<B_ / 16, 128, 0, stream>>>(tokens, W_hh, proj, W_fc, b_fc, out);
}